// Flow_12567074308620
// MI455X (gfx1250) — compile-verified
//
#include <hip/hip_runtime.h>

// Flow-warp (grid_sample bilinear, zeros padding) for code=(16,64,256,256) f32,
// flow=(16,256,256,2) f32.  Memory-bound: ~550MB min traffic -> ~24us at 23.3TB/s.
// One thread = one (b,h,w) pixel x 16 channels; weights/indices amortized,
// stores coalesced + non-temporal, gathers prefetched 2 planes ahead.

constexpr int Bd = 16, Cd = 64, Hd = 256, Wd = 256;
constexpr int HW  = Hd * Wd;          // 65536
constexpr int CG  = 4;                // channel groups per pixel
constexpr int CPT = Cd / CG;          // 16 channels per thread

__global__ __launch_bounds__(256) void flow_warp_kernel(
    const float* __restrict__ code,
    const float* __restrict__ flow,
    float* __restrict__ out)
{
  const int t  = blockIdx.x * 256 + threadIdx.x;
  const int w  = t & (Wd - 1);          // fastest -> coalesced stores
  const int h  = (t >> 8) & (Hd - 1);
  const int cg = (t >> 16) & (CG - 1);
  const int b  = t >> 18;

  // flow offsets (x,y) for this pixel; 8B coalesced read, resides in L2 (8MB)
  const float2 f = ((const float2*)flow)[(b * Hd + h) * Wd + w];

  // identity grid (align_corners=False) + unnormalize, exactly as reference
  const float bx = ((float)w + 0.5f) * (2.0f / Wd) - 1.0f;
  const float by = ((float)h + 0.5f) * (2.0f / Hd) - 1.0f;
  const float gx = f.x + bx;
  const float gy = f.y + by;
  const float ix = ((gx + 1.0f) * (float)Wd - 1.0f) * 0.5f;
  const float iy = ((gy + 1.0f) * (float)Hd - 1.0f) * 0.5f;

  const float x0f = floorf(ix), y0f = floorf(iy);
  const float x1f = x0f + 1.0f, y1f = y0f + 1.0f;

  // bilinear weights (corner pairing matches reference)
  float w00 = (x1f - ix) * (y1f - iy);   // (x0,y0)
  float w01 = (ix - x0f) * (y1f - iy);   // (x1,y0)
  float w10 = (x1f - ix) * (iy - y0f);   // (x0,y1)
  float w11 = (ix - x0f) * (iy - y0f);   // (x1,y1)

  // zeros padding: fold validity into the weights once (branch-free gathers)
  const bool x0v = (x0f >= 0.0f) && (x0f <= (float)(Wd - 1));
  const bool x1v = (x1f >= 0.0f) && (x1f <= (float)(Wd - 1));
  const bool y0v = (y0f >= 0.0f) && (y0f <= (float)(Hd - 1));
  const bool y1v = (y1f >= 0.0f) && (y1f <= (float)(Hd - 1));
  if (!(x0v && y0v)) w00 = 0.0f;
  if (!(x1v && y0v)) w01 = 0.0f;
  if (!(x0v && y1v)) w10 = 0.0f;
  if (!(x1v && y1v)) w11 = 0.0f;

  const int x0 = (int)fminf(fmaxf(x0f, 0.0f), (float)(Wd - 1));
  const int x1 = (int)fminf(fmaxf(x1f, 0.0f), (float)(Wd - 1));
  const int y0 = (int)fminf(fmaxf(y0f, 0.0f), (float)(Hd - 1));
  const int y1 = (int)fminf(fmaxf(y1f, 0.0f), (float)(Hd - 1));

  const int off00 = y0 * Wd + x0;
  const int off01 = y0 * Wd + x1;   // same cacheline as off00 (x adjacency)
  const int off10 = y1 * Wd + x0;
  const int off11 = y1 * Wd + x1;   // same cacheline as off10

  const size_t planeBase = (size_t)(b * Cd + cg * CPT) * HW;
  const float* __restrict__ p = code + planeBase;
  float*       __restrict__ o = out + planeBase + (size_t)(h * Wd + w);

  // warm the first planes' corner rows (global_prefetch_b8, whole cacheline)
  __builtin_prefetch(p + off00, 0, 1);
  __builtin_prefetch(p + off10, 0, 1);
  __builtin_prefetch(p + HW + off00, 0, 1);
  __builtin_prefetch(p + HW + off10, 0, 1);

#pragma unroll
  for (int i = 0; i < CPT; ++i) {
    if (i + 2 < CPT) {
      const float* pn = p + (size_t)(i + 2) * HW;   // 2-plane prefetch distance
      __builtin_prefetch(pn + off00, 0, 1);          // row y0 (covers x0 and x1)
      __builtin_prefetch(pn + off10, 0, 1);          // row y1 (covers x0 and x1)
    }
    const float* pi  = p + (size_t)i * HW;
    const float  v00 = pi[off00];
    const float  v01 = pi[off01];
    const float  v10 = pi[off10];
    const float  v11 = pi[off11];
    const float acc = fmaf(w00, v00, fmaf(w01, v01, fmaf(w10, v10, w11 * v11)));
    // write-once 268MB stream: non-temporal so it doesn't evict code/grid in L2
    __builtin_nontemporal_store(acc, o + (size_t)i * HW);
  }
}

extern "C" void kernel_launch(void* const* d_in, const int* in_sizes, int n_in,
                              void* d_out, int out_size, void* d_ws, size_t ws_size,
                              hipStream_t stream) {
  const float* code = (const float*)d_in[0];   // (16,64,256,256) f32
  const float* flow = (const float*)d_in[1];   // (16,256,256,2)  f32
  float*       out  = (float*)d_out;           // (16,64,256,256) f32
  (void)in_sizes; (void)n_in; (void)out_size; (void)d_ws; (void)ws_size;

  const int total  = Bd * CG * Hd * Wd;        // 4,194,304 threads
  const int blocks = total / 256;              // 16,384 blocks x 8 wave32
  flow_warp_kernel<<<blocks, 256, 0, stream>>>(code, flow, out);
}